// FFN_Experts_48137993453611
// MI455X (gfx1250) — compile-verified
//
#include <hip/hip_runtime.h>
#include <hip/hip_bf16.h>
#include <math.h>

// Problem dims (fixed by setup_inputs in the reference)
constexpr int Bb = 4;
constexpr int Nn = 2048;
constexpr int Dd = 512;
constexpr int Ee = 8;
constexpr int Hh = 2048;
constexpr int ROWS = 16;     // padded M for WMMA (B*k = 8 valid rows)
constexpr int SPLIT = 32;    // n-splits for the xbar reduction
constexpr int WAVES = 8;     // waves per GEMM block
constexpr int CH = 256;      // K-chunk staged in LDS
constexpr int LDH = CH + 4;  // padded LDS row stride: 4 mod 64 -> conflict-free, 16B aligned

typedef __attribute__((ext_vector_type(2))) float v2f;
typedef __attribute__((ext_vector_type(4))) float f32x4;
typedef __attribute__((ext_vector_type(8))) float v8f;
typedef unsigned int u32x4 __attribute__((ext_vector_type(4)));
typedef int i32x8 __attribute__((ext_vector_type(8)));
typedef int i32x4 __attribute__((ext_vector_type(4)));

#if __has_builtin(__builtin_amdgcn_tensor_load_to_lds) && \
    __has_builtin(__builtin_amdgcn_s_wait_tensorcnt)
#define USE_TDM 1
#elif __has_builtin(__builtin_amdgcn_global_load_async_to_lds_b128) && \
      __has_builtin(__builtin_amdgcn_s_wait_asynccnt)
#define USE_ASYNC 1
#endif

__device__ __forceinline__ float gelu_f(float x) {
  float x3 = x * x * x;
  float t = tanhf(0.7978845608028654f * (x + 0.044715f * x3));
  return 0.5f * x * (1.0f + t);
}

// ---- Kernel 1: xbar[b,d] += partial sums over n (xbar zeroed by memsetAsync) ----
__global__ void moe_xbar_accum(const float* __restrict__ x, float* __restrict__ xbar) {
  int b = blockIdx.x / SPLIT;
  int s = blockIdx.x % SPLIT;
  int t = threadIdx.x;  // blockDim.x == Dd/4
  const int rows = Nn / SPLIT;
  const f32x4* xp =
      reinterpret_cast<const f32x4*>(x + ((size_t)b * Nn + (size_t)s * rows) * Dd) + t;
  f32x4 acc = {0.f, 0.f, 0.f, 0.f};
  for (int i = 0; i < rows; ++i) {
    f32x4 v = xp[(size_t)i * (Dd / 4)];
    acc += v;
  }
  float* xb = xbar + b * Dd + t * 4;
  atomicAdd(xb + 0, acc.x);
  atomicAdd(xb + 1, acc.y);
  atomicAdd(xb + 2, acc.z);
  atomicAdd(xb + 3, acc.w);
}

// ---- Kernel 2: build padded A matrix Xk[16 x D] from rows x[b, j, :] ----
__global__ void moe_build_xk(const float* __restrict__ x, const int* __restrict__ kp,
                             float* __restrict__ Xk) {
  int t = blockIdx.x * blockDim.x + threadIdx.x;
  if (t >= ROWS * Dd) return;
  int r = t / Dd, d = t % Dd;
  int kk = *kp; if (kk < 1) kk = 1; if (kk > Ee) kk = Ee;
  float v = 0.f;
  if (r < Bb * kk) {
    int b = r / kk, j = r % kk;  // n = j (reference gathers outs[b, idx, j, :])
    v = x[((size_t)b * Nn + j) * Dd + d];
  }
  Xk[t] = v;
}

// ---- Kernel 3: routing -> softmax over mean scores, top-k values/indices ----
__global__ void moe_routing(const float* __restrict__ xbar, const float* __restrict__ Wr,
                            const float* __restrict__ br, const int* __restrict__ kp,
                            float* __restrict__ vals, int* __restrict__ idxw) {
  __shared__ float sc[Bb * Ee];
  int tid = threadIdx.x;
  int lane = tid & 31, w = tid >> 5;
  for (int p = w; p < Bb * Ee; p += WAVES) {
    int b = p / Ee, e = p % Ee;
    float s = 0.f;
    for (int i = lane; i < Dd; i += 32) s += xbar[b * Dd + i] * Wr[e * Dd + i];
    for (int off = 16; off >= 1; off >>= 1) s += __shfl_xor(s, off, 32);
    if (lane == 0) sc[p] = s * (1.0f / (float)Nn) + br[e];
  }
  __syncthreads();
  if (tid < Bb) {
    int b = tid;
    int kk = *kp; if (kk < 1) kk = 1; if (kk > Ee) kk = Ee;
    float mx = sc[b * Ee];
    for (int e = 1; e < Ee; ++e) mx = fmaxf(mx, sc[b * Ee + e]);
    float ex[Ee]; float sum = 0.f;
    for (int e = 0; e < Ee; ++e) { ex[e] = expf(sc[b * Ee + e] - mx); sum += ex[e]; }
    bool used[Ee];
    for (int e = 0; e < Ee; ++e) used[e] = false;
    for (int j = 0; j < kk; ++j) {
      int best = 0; float bv = -1.0f;
      for (int e = 0; e < Ee; ++e)
        if (!used[e] && ex[e] > bv) { bv = ex[e]; best = e; }
      used[best] = true;
      vals[b * kk + j] = bv / sum;
      idxw[b * kk + j] = best;
    }
  }
}

// ---- Stage a 16 x CH f32 tile (row stride K) from global into LDS (row stride LDH) ----
__device__ __forceinline__ void stage_tile(const float* __restrict__ gsrc, float* sA,
                                           int K, int tid) {
#if defined(USE_TDM)
  // Tensor Data Mover: one DMA descriptor issued by wave 0; hardware applies the
  // LDS row padding (pad_interval=256 DWORDs -> code 7, pad_amount=4 DWORDs -> code 3).
  if (tid < 32) {
    unsigned long long ga = (unsigned long long)(uintptr_t)gsrc;
    unsigned ldsOff = (unsigned)(uintptr_t)(__attribute__((address_space(3))) void*)sA;
    u32x4 g0;
    g0[0] = 1u;                                   // count=1 (valid), user mode
    g0[1] = ldsOff;                               // lds_addr (bytes)
    g0[2] = (unsigned)(ga & 0xFFFFFFFFu);         // global_addr[31:0]
    g0[3] = (unsigned)((ga >> 32) & 0x01FFFFFFu)  // global_addr[56:32]
            | (2u << 30);                         // type = 2 ("image")
    i32x8 g1;
    g1[0] = (2 << 16)      // data_size = 4B
          | (1 << 20)      // pad_enable
          | (7 << 22)      // pad_interval: 256 DWORDs
          | (3 << 25);     // pad_amount: 4 DWORDs
    g1[1] = (K & 0xFFFF) << 16;        // tensor_dim0[15:0] (bits 63:48)
    g1[2] = (ROWS & 0xFFFF) << 16;     // tensor_dim0[31:16]=0 | tensor_dim1[15:0]
    g1[3] = (CH & 0xFFFF) << 16;       // tensor_dim1[31:16]=0 | tile_dim0
    g1[4] = ROWS;                      // tile_dim1 | tile_dim2=0
    g1[5] = K;                         // tensor_dim0_stride[31:0]
    g1[6] = 0;                         // stride0[47:32] | stride1[15:0]
    g1[7] = 0;
    i32x4 gz4 = {0, 0, 0, 0};
    i32x8 gz8 = {0, 0, 0, 0, 0, 0, 0, 0};
    __builtin_amdgcn_tensor_load_to_lds(g0, g1, gz4, gz4, gz8, 0);
    __builtin_amdgcn_s_wait_tensorcnt(0);
  }
#elif defined(USE_ASYNC)
  {
    __attribute__((address_space(3))) float* sl =
        (__attribute__((address_space(3))) float*)sA;
    for (int t = tid; t < ROWS * (CH / 4); t += 256) {
      int m = t / (CH / 4), q4 = t % (CH / 4);
      __builtin_amdgcn_global_load_async_to_lds_b128(
          (__attribute__((address_space(1))) const void*)(gsrc + (size_t)m * K + q4 * 4),
          (__attribute__((address_space(3))) void*)(sl + m * LDH + q4 * 4), 0, 0);
    }
    __builtin_amdgcn_s_wait_asynccnt(0);
  }
#else
  for (int t = tid; t < ROWS * CH; t += 256) {
    int m = t / CH, kq = t % CH;
    sA[m * LDH + kq] = gsrc[(size_t)m * K + kq];
  }
#endif
}

// ---- Kernel 4: per-expert GEMM, M=16, via V_WMMA_F32_16X16X4_F32 ----
// B-matrix element (k,n) = Bw[e*Ncols*K + n*K + k]  (both W1 [E,H,D] and W2 [E,D,H] fit)
// Out[(e*16+m)*Ncols + n]; optional bias + GELU.
__global__ void moe_gemm_experts(const float* __restrict__ A, const float* __restrict__ Bw,
                                 const float* __restrict__ bias, const int* __restrict__ idxw,
                                 const int* __restrict__ kp, float* __restrict__ Out,
                                 int K, int Ncols, int aPerExpert, int doGelu) {
  extern __shared__ float sA[];  // 16 x LDH floats
  int e = blockIdx.x;
  int kk = *kp; if (kk < 1) kk = 1; if (kk > Ee) kk = Ee;
  int nsel = Bb * kk; if (nsel > ROWS) nsel = ROWS;
  bool sel = false;
  for (int t = 0; t < nsel; ++t) sel = sel || (idxw[t] == e);
  if (!sel) return;  // uniform across block

  int tid = threadIdx.x;
  int lane = tid & 31, w = tid >> 5;
  int nl = lane & 15, half = lane >> 4;
  int n0 = blockIdx.y * (16 * WAVES) + w * 16;

  const float* Ae = A + (aPerExpert ? (size_t)e * ROWS * K : (size_t)0);
  const float* Be = Bw + (size_t)e * Ncols * K;

  v8f c = {0.f, 0.f, 0.f, 0.f, 0.f, 0.f, 0.f, 0.f};

  for (int kc = 0; kc < K; kc += CH) {
    __syncthreads();
    stage_tile(Ae + kc, sA, K, tid);
    __syncthreads();
    const float* Brow = Be + (size_t)(n0 + nl) * K + kc + 2 * half;
    const float* Arow = sA + nl * LDH + 2 * half;
    __builtin_prefetch(Brow + CH, 0, 1);  // global_prefetch_b8: next K-chunk of weights
#pragma unroll 8
    for (int k4 = 0; k4 < CH; k4 += 4) {
      // A 16x4 f32: lane m = lane%16; VGPR v, half h -> K = 2h + v
      v2f a; a.x = Arow[k4]; a.y = Arow[k4 + 1];
      // B 4x16 f32: lane n = lane%16; VGPR v, half h -> K = 2h + v
      float2 bb = *reinterpret_cast<const float2*>(Brow + k4);
      v2f bfrag; bfrag.x = bb.x; bfrag.y = bb.y;
      c = __builtin_amdgcn_wmma_f32_16x16x4_f32(false, a, false, bfrag, (short)0, c,
                                                false, false);
    }
  }

  float bv = bias[(size_t)e * Ncols + n0 + nl];
#pragma unroll
  for (int v = 0; v < 8; ++v) {
    int m = 8 * half + v;  // C/D layout: VGPR v, half h -> M = 8h + v, N = lane%16
    float val = c[v] + bv;
    if (doGelu) val = gelu_f(val);
    Out[((size_t)e * ROWS + m) * Ncols + n0 + nl] = val;
  }
}

// ---- Kernel 5: combined[b,d] = sum_j vals[b,j] * Obuf[idx[b,j], b*k+j, d] ----
__global__ void moe_combine(const float* __restrict__ Obuf, const float* __restrict__ vals,
                            const int* __restrict__ idxw, const int* __restrict__ kp,
                            float* __restrict__ comb) {
  int t = blockIdx.x * blockDim.x + threadIdx.x;
  if (t >= Bb * Dd) return;
  int b = t / Dd, d = t % Dd;
  int kk = *kp; if (kk < 1) kk = 1; if (kk > Ee) kk = Ee;
  float s = 0.f;
  for (int j = 0; j < kk; ++j) {
    int r = b * kk + j;
    int e = idxw[r];
    s += vals[r] * Obuf[((size_t)e * ROWS + r) * Dd + d];
  }
  comb[t] = s;
}

// ---- Kernel 6: broadcast combined over the sequence dim, NT float4 stores ----
__global__ void moe_bcast(const float* __restrict__ comb, float* __restrict__ out) {
  size_t t = (size_t)blockIdx.x * blockDim.x + threadIdx.x;  // over B*N*D/4
  const f32x4* c4 = reinterpret_cast<const f32x4*>(comb);
  f32x4* o4 = reinterpret_cast<f32x4*>(out);
  size_t d4 = t % (Dd / 4);
  size_t bn = t / (Dd / 4);
  size_t b = bn / Nn;
  f32x4 v = c4[b * (Dd / 4) + d4];
  __builtin_nontemporal_store(v, o4 + t);  // write-once stream: keep L2 for weights/x
}

extern "C" void kernel_launch(void* const* d_in, const int* in_sizes, int n_in,
                              void* d_out, int out_size, void* d_ws, size_t ws_size,
                              hipStream_t stream) {
  const float* x  = (const float*)d_in[0];
  const float* Wr = (const float*)d_in[1];
  const float* br = (const float*)d_in[2];
  const float* W1 = (const float*)d_in[3];
  const float* b1 = (const float*)d_in[4];
  const float* W2 = (const float*)d_in[5];
  const float* b2 = (const float*)d_in[6];
  const int*   kp = (const int*)d_in[7];
  float* out = (float*)d_out;

  // Workspace layout (float offsets; all tiles stay 16B-aligned)
  float* ws    = (float*)d_ws;
  float* xbar  = ws;                       // B*D   (reused as `comb` after routing)
  float* vals  = xbar + Bb * Dd;           // ROWS
  int*   idxw  = (int*)(vals + ROWS);      // ROWS
  float* Xk    = (float*)(idxw + ROWS);    // ROWS*D
  float* Hbuf  = Xk + ROWS * Dd;           // E*ROWS*H
  float* Obuf  = Hbuf + (size_t)Ee * ROWS * Hh;  // E*ROWS*D

  (void)hipMemsetAsync(xbar, 0, (size_t)Bb * Dd * sizeof(float), stream);

  moe_xbar_accum<<<Bb * SPLIT, Dd / 4, 0, stream>>>(x, xbar);
  moe_build_xk<<<(ROWS * Dd + 255) / 256, 256, 0, stream>>>(x, kp, Xk);
  moe_routing<<<1, 256, 0, stream>>>(xbar, Wr, br, kp, vals, idxw);

  size_t ldsBytes = (size_t)ROWS * LDH * sizeof(float);
  // GEMM1: [16 x D] * W1[e]^T -> Hbuf [E,16,H], bias b1 + GELU
  moe_gemm_experts<<<dim3(Ee, Hh / (16 * WAVES)), 32 * WAVES, ldsBytes, stream>>>(
      Xk, W1, b1, idxw, kp, Hbuf, Dd, Hh, 0, 1);
  // GEMM2: Hbuf[e] [16 x H] * W2[e]^T -> Obuf [E,16,D], bias b2
  moe_gemm_experts<<<dim3(Ee, Dd / (16 * WAVES)), 32 * WAVES, ldsBytes, stream>>>(
      Hbuf, W2, b2, idxw, kp, Obuf, Hh, Dd, 1, 0);

  moe_combine<<<(Bb * Dd + 255) / 256, 256, 0, stream>>>(Obuf, vals, idxw, kp, xbar);
  moe_bcast<<<(size_t)Bb * Nn * Dd / 4 / 256, 256, 0, stream>>>(xbar, out);
}